// SupportLowRankRNN_58145267253494
// MI455X (gfx1250) — compile-verified
//
#include <hip/hip_runtime.h>
#include <cmath>

#define I_DIM 4
#define H_DIM 2048
#define O_DIM 2
#define S_DIM 2
#define GB_DIM 8
#define B_DIM 32
#define T_DIM 1024
#define NOISE_STD_C 0.05f
#define ALPHA_C 0.2f

// LDS layout (floats): u tile [0,32768) | noise 2x16384 [32768,65536) | red [65536,65600)
#define U_BASE_F 0
#define NZ_BASE_F 32768
#define NZ_BASE_BYTES (NZ_BASE_F * 4)
#define NZ_BUF_BYTES 65536
#define RED_BASE_F 65536
#define SMEM_FLOATS (RED_BASE_F + 64)

typedef __attribute__((ext_vector_type(2))) float v2f;
typedef __attribute__((ext_vector_type(4))) float v4f;
typedef __attribute__((ext_vector_type(8))) float v8f;

#if defined(__gfx1250__) && __has_builtin(__builtin_amdgcn_tanhf)
#define TANHF(x) __builtin_amdgcn_tanhf(x)
#else
#define TANHF(x) tanhf(x)
#endif

// Async copy 16B global -> LDS (ASYNCcnt path), non-temporal: noise is
// read exactly once, keep it out of the 192MB L2's working set.
__device__ __forceinline__ void async_copy_b128_nt(unsigned lds_off, const void* gaddr) {
  asm volatile("global_load_async_to_lds_b128 %0, %1, off th:TH_LOAD_NT"
               :: "v"(lds_off), "v"(gaddr)
               : "memory");
}

template <int N>
__device__ __forceinline__ void wait_asynccnt() {
#if defined(__gfx1250__) && __has_builtin(__builtin_amdgcn_s_wait_asynccnt)
  __builtin_amdgcn_s_wait_asynccnt(N);
#else
  asm volatile("s_wait_asynccnt %0" :: "i"(N) : "memory");
#endif
}

// Issue one 8-step noise half-block (8 x H floats = 64KB) into LDS buffer.
__device__ __forceinline__ void issue_noise_half(const float* __restrict__ nz_b,
                                                 int tstart, unsigned lds_byte_base,
                                                 int tid) {
#pragma unroll
  for (int i = 0; i < 16; ++i) {
    const float* g = nz_b + (size_t)tstart * H_DIM + i * 1024 + tid * 4;
    async_copy_b128_nt(lds_byte_base + (unsigned)(i * 4096 + tid * 16), (const void*)g);
  }
}

// val = sum_s sup[s,h] * ( sum_g w[row,s,g]*basis[g] + bias[row,s] )
__device__ __forceinline__ float proxy_coef(const float* __restrict__ w,
                                            const float* __restrict__ bias,
                                            const float* basis, const float* supv,
                                            int row) {
  float acc = 0.f;
#pragma unroll
  for (int s = 0; s < S_DIM; ++s) {
    float t = bias ? bias[row * S_DIM + s] : 0.f;
#pragma unroll
    for (int g = 0; g < GB_DIM; ++g)
      t = fmaf(w[(row * S_DIM + s) * GB_DIM + g], basis[g], t);
    acc = fmaf(supv[s], t, acc);
  }
  return acc;
}

__device__ __forceinline__ void wave_reduce4(float& a, float& b, float& c, float& d) {
#pragma unroll
  for (int off = 16; off > 0; off >>= 1) {
    a += __shfl_xor(a, off, 32);
    b += __shfl_xor(b, off, 32);
    c += __shfl_xor(c, off, 32);
    d += __shfl_xor(d, off, 32);
  }
}

__global__ void __launch_bounds__(256)
support_lowrank_rnn_kernel(const float* __restrict__ input,   // (B,T,I)
                           const float* __restrict__ noise,   // (B,T,H)
                           const float* __restrict__ gb,      // (GB,H)
                           const float* __restrict__ sup,     // (S,H)
                           const float* __restrict__ wi_w,    // (I,S,GB)
                           const float* __restrict__ m_w,     // (R,S,GB)
                           const float* __restrict__ n_w,     // (R,S,GB)
                           const float* __restrict__ wo_w,    // (O,S,GB)
                           const float* __restrict__ wi_b,    // (I,S)
                           const float* __restrict__ m_b,     // (R,S)
                           const float* __restrict__ n_b,     // (R,S)
                           const float* __restrict__ h0_w,    // (S,GB)
                           float* __restrict__ out,           // (B,T,O)
                           float* __restrict__ traj) {        // (B,T,H)
  extern __shared__ float smem[];
  float* u_smem = smem + U_BASE_F;   // 16 * H_DIM
  float* nz_smem = smem + NZ_BASE_F; // 2 * 8 * H_DIM
  float* red = smem + RED_BASE_F;    // 2 buffers * 8 waves * 4

  const int b = blockIdx.x;
  const int tid = (int)threadIdx.x;
  const int lane = tid & 31;
  const int wave = tid >> 5;
  const int lhalf = (lane < 16) ? 0 : 1;  // half-wave: K=0/1 vs K=2/3
  const int lmod = lane & 15;
  const float invH = 1.0f / (float)H_DIM;

  const float* inp_b = input + (size_t)b * T_DIM * I_DIM;
  const float* nz_b = noise + (size_t)b * T_DIM * H_DIM;
  float* traj_b = traj + (size_t)b * T_DIM * H_DIM;
  float* out_b = out + (size_t)b * T_DIM * O_DIM;

  // Kick off the async noise stream for steps 0..7 immediately.
  issue_noise_half(nz_b, 0, NZ_BASE_BYTES, tid);

  // ---------------- per-thread RNN columns (8 contiguous h each) -------------
  const int c0 = tid * 8;
  float m0[8], m1[8], n0[8], n1[8], w0[8], w1[8], h[8];
  float pn0 = 0.f, pn1 = 0.f, po0 = 0.f, po1 = 0.f;
#pragma unroll
  for (int j = 0; j < 8; ++j) {
    const int hc = c0 + j;
    float basis[GB_DIM], supv[S_DIM];
#pragma unroll
    for (int g = 0; g < GB_DIM; ++g) basis[g] = gb[g * H_DIM + hc];
#pragma unroll
    for (int s = 0; s < S_DIM; ++s) supv[s] = sup[s * H_DIM + hc];
    m0[j] = proxy_coef(m_w, m_b, basis, supv, 0);
    m1[j] = proxy_coef(m_w, m_b, basis, supv, 1);
    n0[j] = proxy_coef(n_w, n_b, basis, supv, 0);
    n1[j] = proxy_coef(n_w, n_b, basis, supv, 1);
    w0[j] = proxy_coef(wo_w, nullptr, basis, supv, 0);
    w1[j] = proxy_coef(wo_w, nullptr, basis, supv, 1);
    h[j] = proxy_coef(h0_w, nullptr, basis, supv, 0);  // h0, broadcast over batch
    float t0h = TANHF(h[j]);                           // r for step 0
    pn0 = fmaf(t0h, n0[j], pn0);
    pn1 = fmaf(t0h, n1[j], pn1);
  }

  // ------------- per-lane WMMA B operands: wi columns for this wave ----------
  // B (4x16 f32): VGPR0 = K={0|2} row, VGPR1 = K={1|3}, N = lane%16 column.
  const int k0 = lhalf ? 2 : 0;
  v2f breg[16];
#pragma unroll
  for (int jt = 0; jt < 16; ++jt) {
    const int col = wave * 256 + jt * 16 + lmod;
    float basis[GB_DIM], supv[S_DIM];
#pragma unroll
    for (int g = 0; g < GB_DIM; ++g) basis[g] = gb[g * H_DIM + col];
#pragma unroll
    for (int s = 0; s < S_DIM; ++s) supv[s] = sup[s * H_DIM + col];
    breg[jt].x = proxy_coef(wi_w, wi_b, basis, supv, k0);
    breg[jt].y = proxy_coef(wi_w, wi_b, basis, supv, k0 + 1);
  }

  // Prologue reduction partials for t=0 (po == out[-1], unused -> 0)
  wave_reduce4(pn0, pn1, po0, po1);
  if (lane == 0) {
    float* wbuf = red + 0 * 32 + wave * 4;
    wbuf[0] = pn0; wbuf[1] = pn1; wbuf[2] = 0.f; wbuf[3] = 0.f;
  }

  // A-tile for the first WMMA group, software-pipelined one group ahead.
  // A (16x4 f32): lanes 0-15 -> M=lane, K=0/1; lanes 16-31 -> M=lane-16, K=2/3
  v2f a_cur = *(const v2f*)(inp_b + lmod * I_DIM + k0);

  for (int hi = 0; hi < T_DIM / 8; ++hi) {
    const int t0 = hi * 8;

    if ((hi & 1) == 0) {
      // ----- WMMA u-tile: u[t0..t0+15][:] = input_tile(16x4) @ wi(4xH) -------
#pragma unroll
      for (int jt = 0; jt < 16; ++jt) {
        v8f cz = {};
        v8f d = __builtin_amdgcn_wmma_f32_16x16x4_f32(
            false, a_cur, false, breg[jt], (short)0, cz, false, false);
        const int col = wave * 256 + jt * 16 + lmod;
#pragma unroll
        for (int r = 0; r < 8; ++r)  // D: VGPR r -> M=r (lanes<16) / M=r+8
          u_smem[(r + lhalf * 8) * H_DIM + col] = d[r];
      }
      // Prefetch next group's A operand (consumed 16 steps from now).
      const int tnx = (t0 + 16 < T_DIM) ? t0 + 16 : t0;
      a_cur = *(const v2f*)(inp_b + (tnx + lmod) * I_DIM + k0);
    }

    // Issue the next 8-step noise half, then retire the current one.
    if (hi + 1 < T_DIM / 8) {
      issue_noise_half(nz_b, (hi + 1) * 8,
                       NZ_BASE_BYTES + (unsigned)(((hi + 1) & 1) * NZ_BUF_BYTES), tid);
      wait_asynccnt<16>();  // 16 newer ops in flight; older half complete
    } else {
      wait_asynccnt<0>();
    }
    __syncthreads();  // noise half + u tile (on even hi) visible to all waves

    const float* nz_half = nz_smem + (hi & 1) * (8 * H_DIM);

    for (int tl = 0; tl < 8; ++tl) {
      const int t = t0 + tl;

      // Broadcast reduction results written at end of previous step.
      float Sn0 = 0.f, Sn1 = 0.f, So0 = 0.f, So1 = 0.f;
      const v4f* rb = (const v4f*)(red + (t & 1) * 32);
#pragma unroll
      for (int wv = 0; wv < 8; ++wv) {
        v4f rv = rb[wv];
        Sn0 += rv.x; Sn1 += rv.y; So0 += rv.z; So1 += rv.w;
      }
      if (t > 0 && tid == 0) {
        out_b[(t - 1) * O_DIM + 0] = So0 * invH;
        out_b[(t - 1) * O_DIM + 1] = So1 * invH;
      }

      // Noise from LDS (written by async DMA), u from WMMA tile.
      const float* nzp = nz_half + tl * H_DIM + c0;
      const v4f nzA = *(const v4f*)nzp;
      const v4f nzB = *(const v4f*)(nzp + 4);
      const int tl16 = t & 15;

      float qn0 = 0.f, qn1 = 0.f, qo0 = 0.f, qo1 = 0.f;
#pragma unroll
      for (int j = 0; j < 8; ++j) {
        const float nz = (j < 4) ? nzA[j] : nzB[j - 4];
        const float u = u_smem[tl16 * H_DIM + c0 + j];
        const float rec = (m0[j] * Sn0 + m1[j] * Sn1) * invH;
        float hv = h[j];
        hv = hv + NOISE_STD_C * nz + ALPHA_C * (-hv + rec + u);
        h[j] = hv;
        const float r = TANHF(hv);
        qn0 = fmaf(r, n0[j], qn0);
        qn1 = fmaf(r, n1[j], qn1);
        qo0 = fmaf(r, w0[j], qo0);
        qo1 = fmaf(r, w1[j], qo1);
      }

      // Trajectory store: 2x b128 per thread, non-temporal (write-once stream,
      // never re-read -> keep it from evicting the L2 working set).
      v4f s0, s1;
      s0.x = h[0]; s0.y = h[1]; s0.z = h[2]; s0.w = h[3];
      s1.x = h[4]; s1.y = h[5]; s1.z = h[6]; s1.w = h[7];
      __builtin_nontemporal_store(s0, (v4f*)(traj_b + (size_t)t * H_DIM + c0));
      __builtin_nontemporal_store(s1, (v4f*)(traj_b + (size_t)t * H_DIM + c0 + 4));

      wave_reduce4(qn0, qn1, qo0, qo1);
      if (lane == 0) {
        float* wbuf = red + ((t + 1) & 1) * 32 + wave * 4;
        wbuf[0] = qn0; wbuf[1] = qn1; wbuf[2] = qo0; wbuf[3] = qo1;
      }
      __syncthreads();  // one barrier per step
    }
  }

  // Final output (t = T-1): partials live in buffer (T & 1) == 0.
  {
    float So0 = 0.f, So1 = 0.f;
    const v4f* rb = (const v4f*)(red + (T_DIM & 1) * 32);
#pragma unroll
    for (int wv = 0; wv < 8; ++wv) {
      v4f rv = rb[wv];
      So0 += rv.z; So1 += rv.w;
    }
    if (tid == 0) {
      out_b[(T_DIM - 1) * O_DIM + 0] = So0 * invH;
      out_b[(T_DIM - 1) * O_DIM + 1] = So1 * invH;
    }
  }
}

extern "C" void kernel_launch(void* const* d_in, const int* in_sizes, int n_in,
                              void* d_out, int out_size, void* d_ws, size_t ws_size,
                              hipStream_t stream) {
  (void)in_sizes; (void)n_in; (void)out_size; (void)d_ws; (void)ws_size;
  const float* input = (const float*)d_in[0];
  const float* noise = (const float*)d_in[1];
  const float* gb    = (const float*)d_in[2];
  const float* sup   = (const float*)d_in[3];
  const float* wi_w  = (const float*)d_in[4];
  const float* m_w   = (const float*)d_in[5];
  const float* n_w   = (const float*)d_in[6];
  const float* wo_w  = (const float*)d_in[7];
  const float* wi_b  = (const float*)d_in[8];
  const float* m_b   = (const float*)d_in[9];
  const float* n_b   = (const float*)d_in[10];
  const float* h0_w  = (const float*)d_in[11];

  float* out = (float*)d_out;                          // (B,T,O)
  float* traj = out + (size_t)B_DIM * T_DIM * O_DIM;   // (B,T,H)

  const size_t smem_bytes = (size_t)SMEM_FLOATS * sizeof(float);  // 262,400 B
  support_lowrank_rnn_kernel<<<dim3(B_DIM), dim3(256), smem_bytes, stream>>>(
      input, noise, gb, sup, wi_w, m_w, n_w, wo_w, wi_b, m_b, n_b, h0_w,
      out, traj);
}